// Loss_76338748719987
// MI455X (gfx1250) — compile-verified
//
#include <hip/hip_runtime.h>
#include <hip/hip_bf16.h>

// Problem constants (fixed by the reference's setup_inputs)
#define BB 32
#define LL 256
#define TT 2048
#define NEG_PAD (-1.0e12f)
#define INV_2SIG2 3.125f   // 1 / (2 * 0.4^2)

typedef float v2f __attribute__((ext_vector_type(2)));
typedef float v8f __attribute__((ext_vector_type(8)));

// ---------------------------------------------------------------------------
// Wave32 sum reduction via the matrix unit: D = A(16x4) * ones(4x16).
// Each lane contributes one f32 in A; B = all-ones makes the exact A-layout
// irrelevant (row sums == sum of that lane-group's contributions, exact in f32).
// After WMMA, lanes 0-15 hold sum(rows 0-7), lanes 16-31 hold sum(rows 8-15);
// one xor-16 shuffle completes the 32-lane total (broadcast to all lanes).
// ---------------------------------------------------------------------------
__device__ inline float wave_sum_wmma(float v) {
    v2f a;  a[0] = v;    a[1] = 0.0f;
    v2f bo; bo[0] = 1.0f; bo[1] = 1.0f;
    v8f c = {};
    v8f d = __builtin_amdgcn_wmma_f32_16x16x4_f32(
        /*neg_a=*/false, a, /*neg_b=*/false, bo,
        /*c_mod=*/(short)0, c, /*reuse_a=*/false, /*reuse_b=*/false);
    float s = d[0] + d[1] + d[2] + d[3] + d[4] + d[5] + d[6] + d[7];
    s += __shfl_xor(s, 16, 32);
    return s;
}

// Block (256-thread) reduction: WMMA per wave, then LDS across 8 waves.
// Returns the total on thread 0 (others: undefined).
__device__ inline float block_sum(float v) {
    __shared__ float sm[8];
    float ws = wave_sum_wmma(v);             // uniform across the wave
    const int wave = threadIdx.x >> 5;
    if ((threadIdx.x & 31) == 0) sm[wave] = ws;
    __syncthreads();
    float tot = 0.0f;
    if (threadIdx.x == 0) {
        #pragma unroll
        for (int w = 0; w < 8; ++w) tot += sm[w];
    }
    return tot;
}

// ---------------------------------------------------------------------------
// Kernel 1: guided-attention partial sums. One block per (t, b); 256 threads
// cover l = 0..255 with a contiguous 1 KB read of att_ws.
// ---------------------------------------------------------------------------
__global__ __launch_bounds__(256) void ga_partial(
    const float* __restrict__ att, const int* __restrict__ tl,
    const int* __restrict__ ml, float* __restrict__ part) {
    const int t = blockIdx.x, b = blockIdx.y, l = threadIdx.x;
    const int ili = tl[b], oli = ml[b];
    float v = 0.0f;
    if (t < oli && l < ili) {
        const float il = (float)ili, ol = (float)oli;
        const float dd = (float)l / il - (float)t / ol;
        const float w = 1.0f - __expf(-dd * dd * INV_2SIG2);
        v = w * att[((size_t)b * TT + t) * LL + l];
    }
    const float tot = block_sum(v);
    if (threadIdx.x == 0) part[(size_t)b * TT + t] = tot;
}

// ---------------------------------------------------------------------------
// Kernel 2: fold B*T partials into the scalar loss (deterministic order).
// ---------------------------------------------------------------------------
__global__ __launch_bounds__(256) void ga_final(
    const float* __restrict__ part, const int* __restrict__ ml,
    float* __restrict__ out) {
    float v = 0.0f;
    for (int k = threadIdx.x; k < BB * TT; k += 256) v += part[k];
    const float tot = block_sum(v);
    if (threadIdx.x == 0) {
        int den = 0;
        #pragma unroll
        for (int b = 0; b < BB; ++b) den += ml[b];
        out[0] = tot / (float)den;
    }
}

// ---------------------------------------------------------------------------
// Kernel 3: Viterbi forward DP. One wave32 per batch; lane owns 8 contiguous
// text positions (l = 8*lane + j) in registers. Instead of storing log_beta
// (64 MB), emit the backtrack decision bits directly:
//   G[b][t] bit rr = ( beta[t][(rr-1) mod 256] > beta[t][rr] )
// packed one byte per lane (rr = 8*lane .. 8*lane+7) -> 32 B per column.
// ---------------------------------------------------------------------------
__global__ __launch_bounds__(32) void viterbi_fwd(
    const float* __restrict__ logp, unsigned char* __restrict__ G) {
    const int b = blockIdx.x, lane = threadIdx.x;
    const float* lp = logp + (size_t)b * LL * TT;   // [l*TT + t]
    unsigned char* gb = G + (size_t)b * TT * 32;

    float prev[8];
    float topw = NEG_PAD;   // prev[7] of lane-1 (wrapped), refreshed each step

    for (int k = 0; k < TT / 4; ++k) {
        float4 q[8];
        #pragma unroll
        for (int j = 0; j < 8; ++j)
            q[j] = *(const float4*)(lp + (size_t)(lane * 8 + j) * TT + 4 * k);

        #pragma unroll
        for (int u = 0; u < 4; ++u) {
            const int t = 4 * k + u;
            float lpv[8];
            #pragma unroll
            for (int j = 0; j < 8; ++j)
                lpv[j] = (u == 0) ? q[j].x : (u == 1) ? q[j].y
                       : (u == 2) ? q[j].z : q[j].w;

            if (t == 0) {
                #pragma unroll
                for (int j = 0; j < 8; ++j) prev[j] = NEG_PAD;
                if (lane == 0) prev[0] = lpv[0];
            } else {
                const float sh0 = (lane == 0) ? NEG_PAD : topw;
                float nxt[8];
                nxt[0] = fmaxf(prev[0], sh0) + lpv[0];
                #pragma unroll
                for (int j = 1; j < 8; ++j)
                    nxt[j] = fmaxf(prev[j], prev[j - 1]) + lpv[j];
                #pragma unroll
                for (int j = 0; j < 8; ++j) prev[j] = nxt[j];
            }

            // wrapped neighbor (lane 0 receives lane 31 -> index 255)
            topw = __shfl(prev[7], (lane + 31) & 31, 32);

            unsigned int bits = (topw > prev[0]) ? 1u : 0u;
            #pragma unroll
            for (int j = 1; j < 8; ++j)
                bits |= ((prev[j - 1] > prev[j]) ? 1u : 0u) << j;
            gb[(size_t)t * 32 + lane] = (unsigned char)bits;
        }
    }
}

// ---------------------------------------------------------------------------
// Kernel 4: backtracking. One wave; lane = batch. The G column index is a
// deterministic function of the step (independent of the data-dependent row),
// so loads are software-pipelined 4 deep; only ~16 ALU ops stay on the
// loop-carried dependency chain.
// ---------------------------------------------------------------------------
__device__ inline int bt_col(int mlb, int i) {
    int c = mlb - 1 - i;
    if (c < -1) c = -1;          // cols = max(cols,0)-1 saturates at -1
    int cm1 = c - 1;             // python-style mod T (c-1 >= -2)
    if (cm1 < 0) cm1 += TT;
    return cm1;
}

__global__ __launch_bounds__(32) void viterbi_bwd(
    const unsigned char* __restrict__ G, const int* __restrict__ tl,
    const int* __restrict__ ml, float* __restrict__ path) {
    const int b = threadIdx.x;
    const int mlb = ml[b];
    float rows = (float)(tl[b] - 1);
    path[(size_t)b * TT + (TT - 1)] = rows;

    const uint4* gb = (const uint4*)(G + (size_t)b * TT * 32);  // 2 x uint4 per column
    const int NS = TT - 1;  // 2047 backtrack steps

    uint4 w0[4], w1[4];
    #pragma unroll
    for (int u = 0; u < 4; ++u) {
        const int ci = bt_col(mlb, u);
        w0[u] = gb[2 * ci];
        w1[u] = gb[2 * ci + 1];
    }

    for (int blk = 0; blk < NS; blk += 4) {
        #pragma unroll
        for (int u = 0; u < 4; ++u) {
            const int i = blk + u;
            if (i < NS) {
                const int r  = (int)rows;        // rows in {-1..255}, integral
                const int rr = r & (LL - 1);     // python mod 256
                // 8-way word select by rr>>5 (cndmask tree, no memory dep)
                const int d = rr >> 5;
                const uint4 g = (d & 4) ? w1[u] : w0[u];
                const unsigned p0 = (d & 2) ? g.z : g.x;
                const unsigned p1 = (d & 2) ? g.w : g.y;
                const unsigned word = (d & 1) ? p1 : p0;
                const int is_go = (int)((word >> (rr & 31)) & 1u);

                rows = fmaxf(rows - (float)is_go + 1.0f, 0.0f) - 1.0f;
                path[(size_t)b * TT + (TT - 2 - i)] = rows;

                // refill slot u with the column for step i+4 (clamped, harmless)
                int ni = i + 4; if (ni >= NS) ni = NS - 1;
                const int ci = bt_col(mlb, ni);
                w0[u] = gb[2 * ci];
                w1[u] = gb[2 * ci + 1];
            }
        }
    }
}

// ---------------------------------------------------------------------------
// Kernel 5: expand path -> shifted one-hot alignment (B, L, T).
// out[b,l,t] = (t + pad < T) && (path[b, t+pad] == l), pad = T - mel_len[b].
// ---------------------------------------------------------------------------
__global__ __launch_bounds__(256) void build_align(
    const float* __restrict__ path, const int* __restrict__ ml,
    float* __restrict__ out) {
    const int l = blockIdx.x, b = blockIdx.y;
    const int pad = TT - ml[b];
    const float lf = (float)l;
    float* o = out + 1 + ((size_t)b * LL + l) * TT;
    #pragma unroll
    for (int j = 0; j < TT / 256; ++j) {
        const int t = threadIdx.x + 256 * j;
        const int tp = t + pad;
        float v = 0.0f;
        if (tp < TT) v = (path[(size_t)b * TT + tp] == lf) ? 1.0f : 0.0f;
        o[t] = v;
    }
}

// ---------------------------------------------------------------------------
extern "C" void kernel_launch(void* const* d_in, const int* in_sizes, int n_in,
                              void* d_out, int out_size, void* d_ws, size_t ws_size,
                              hipStream_t stream) {
    (void)in_sizes; (void)n_in; (void)out_size; (void)ws_size;
    const float* logp = (const float*)d_in[0];   // (B, L, T) f32
    const float* att  = (const float*)d_in[1];   // (B, T, L) f32
    const int*   tl   = (const int*)d_in[2];     // (B,) i32
    const int*   ml   = (const int*)d_in[3];     // (B,) i32
    float* out = (float*)d_out;                  // [0]=loss, [1..]=align (B,L,T)

    char* ws = (char*)d_ws;
    unsigned char* G = (unsigned char*)ws;                          // 2 MB
    float* path = (float*)(ws + (size_t)BB * TT * 32);              // 256 KB
    float* part = (float*)(ws + (size_t)BB * TT * 32
                              + (size_t)BB * TT * sizeof(float));   // 256 KB

    ga_partial <<<dim3(TT, BB), 256, 0, stream>>>(att, tl, ml, part);
    ga_final   <<<1,            256, 0, stream>>>(part, ml, out);
    viterbi_fwd<<<BB,            32, 0, stream>>>(logp, G);
    viterbi_bwd<<<1,             32, 0, stream>>>(G, tl, ml, path);
    build_align<<<dim3(LL, BB), 256, 0, stream>>>(path, ml, out);
}